// GuidedBlur_61684320305790
// MI455X (gfx1250) — compile-verified
//
#include <hip/hip_runtime.h>

// GuidedBlur on MI455X (gfx1250), f32, two fused tile kernels.
// Both separable 7-tap passes run as banded-matrix WMMA:
//   hblur: H(22x16) = Q(22x24) x S_h(24x16)   (A = data, B = 0/1 band)
//   vblur: D(16x16) = S_v(16x24) x H(24x16)   (A = 0/1 band, B = data)
// with V_WMMA_F32_16X16X4_F32 chains (18 WMMAs per channel).

typedef float v2f __attribute__((ext_vector_type(2)));
typedef float v8f __attribute__((ext_vector_type(8)));

#if __has_builtin(__builtin_amdgcn_wmma_f32_16x16x4_f32)
#define GF_HAVE_WMMA 1
#else
#define GF_HAVE_WMMA 0
#endif

namespace {
constexpr int HH  = 512;
constexpr int WW  = 512;
constexpr int NB  = 4;     // batch
constexpr int IT  = 22;    // 16 + 2*3 halo rows
constexpr int TS  = 25;    // raw-tile row stride (gcd(25,64)=1 -> no bank conflicts)
constexpr float INV49 = 1.0f / 49.0f;
constexpr float EPS   = 0.01f;
}

__device__ __forceinline__ int refl(int i) {
  if (i < 0) i = -i;                 // jnp 'reflect': -1 -> 1
  if (i >= WW) i = 2 * WW - 2 - i;   // 512 -> 510
  return i;
}

// 0/1 band: 1 iff n <= k <= n+6 (7-tap window), zero for the K-pad rows.
__device__ __forceinline__ float band(int n, int k) {
  return (k >= n && k <= n + 6) ? 1.0f : 0.0f;
}

#if GF_HAVE_WMMA
__device__ __forceinline__ v8f wmma4(v2f a, v2f b, v8f c) {
  return __builtin_amdgcn_wmma_f32_16x16x4_f32(
      /*neg_a=*/false, a, /*neg_b=*/false, b,
      /*c_mod=*/(short)0, c, /*reuse_a=*/false, /*reuse_b=*/false);
}
#endif

// Full 7x7 blur of channel q (q = Aq[.]* (Bq? Bq[.]:1), 22x24 tile, stride TS)
// into sSdst[256], using per-wave sH (24x16, rows 22/23 pre-zeroed).
// colM = lane&15, half = lane>>4.
__device__ __forceinline__ void blur7x7(const float* Aq, const float* Bq,
                                        float (*sH)[16], float* sSdst,
                                        int colM, int half, int lane) {
#if GF_HAVE_WMMA
  // -------- horizontal: H = Q x S_h ----------
#pragma unroll
  for (int rb = 0; rb < 32; rb += 16) {
    int m = rb + colM;
    if (m > 21) m = 21;                       // clamp pad rows (discarded)
    const float* qa = Aq + m * TS;
    const float* qb = Bq ? Bq + m * TS : nullptr;
    v8f acc = {};
#pragma unroll
    for (int k0 = 0; k0 < 24; k0 += 4) {
      const int ka0 = k0 + 2 * half;          // K rows held by this lane-half
      const int ka1 = ka0 + 1;
      v2f a, b;
      a.x = qa[ka0];
      a.y = qa[ka1];
      if (qb) { a.x *= qb[ka0]; a.y *= qb[ka1]; }
      b.x = band(colM, ka0);
      b.y = band(colM, ka1);
      acc = wmma4(a, b, acc);
    }
#pragma unroll
    for (int j = 0; j < 8; ++j) {
      const int r = rb + j + 8 * half;        // D: VGPR j -> M = j + 8*half
      if (r < IT) sH[r][colM] = acc[j];
    }
  }
  // -------- vertical: D = S_v x H ----------
  {
    v8f acc = {};
#pragma unroll
    for (int k0 = 0; k0 < 24; k0 += 4) {
      const int ka0 = k0 + 2 * half;
      const int ka1 = ka0 + 1;
      v2f a, b;
      a.x = band(colM, ka0);
      a.y = band(colM, ka1);
      b.x = sH[ka0][colM];
      b.y = sH[ka1][colM];
      acc = wmma4(a, b, acc);
    }
#pragma unroll
    for (int j = 0; j < 8; ++j)
      sSdst[(j + 8 * half) * 16 + colM] = acc[j] * INV49;
  }
#else
  // VALU fallback
  for (int idx = lane; idx < IT * 16; idx += 32) {
    const int r = idx >> 4, c = idx & 15;
    float s = 0.0f;
#pragma unroll
    for (int d = 0; d < 7; ++d) {
      float q = Aq[r * TS + c + d];
      if (Bq) q *= Bq[r * TS + c + d];
      s += q;
    }
    sH[r][c] = s;
  }
  for (int idx = lane; idx < 256; idx += 32) {
    const int y = idx >> 4, x = idx & 15;
    float s = 0.0f;
#pragma unroll
    for (int d = 0; d < 7; ++d) s += sH[y + d][x];
    sSdst[idx] = s * INV49;
  }
#endif
}

// ---------------- Kernel 1: 25 blurred stats + 3x3 solve -> a(12),b(4) -----
__global__ __launch_bounds__(256) void gf_stats_kernel(
    const float* __restrict__ G, const float* __restrict__ P,
    float* __restrict__ ab) {
  __shared__ float sG[3][IT][TS];
  __shared__ float sP[4][IT][TS];
  __shared__ float sH[8][24][16];   // per-wave H buffer (rows 22/23 zero)
  __shared__ float sS[25][256];     // blurred stats per tile pixel

  const int tid  = threadIdx.x;
  const int wave = __builtin_amdgcn_readfirstlane(tid >> 5);  // scalarize
  const int lane = tid & 31;
  const int colM = lane & 15, half = lane >> 4;
  const int x0 = blockIdx.x * 16, y0 = blockIdx.y * 16, batch = blockIdx.z;

  // Load guidance(3)+input(4) tiles with reflect halo; zero-fill K-pad cols.
  for (int idx = tid; idx < 7 * IT * 24; idx += 256) {
    const int ch  = idx / (IT * 24);
    const int rem = idx - ch * (IT * 24);
    const int r = rem / 24, c = rem - r * 24;
    float v = 0.0f;
    if (c < IT) {
      const int gy = refl(y0 - 3 + r), gx = refl(x0 - 3 + c);
      v = (ch < 3) ? G[((batch * 3 + ch) * HH + gy) * WW + gx]
                   : P[((batch * 4 + (ch - 3)) * HH + gy) * WW + gx];
    }
    if (ch < 3) sG[ch][r][c] = v;
    else        sP[ch - 3][r][c] = v;
  }
  sH[wave][22 + half][colM] = 0.0f;   // zero K-pad rows of H once
  __syncthreads();

  // 25 channels: 0-2 I, 3-8 I*I (sym), 9-12 p, 13-24 I*p   (scalar ch loop)
  for (int ch = wave; ch < 25; ch += 8) {
    const float *Aq, *Bq;
    if (ch < 3)         { Aq = &sG[ch][0][0];     Bq = nullptr; }
    else if (ch < 9)    {
      const int t = ch - 3;
      const int pa = (t < 3) ? 0 : ((t < 5) ? 1 : 2);
      const int pb = (t < 3) ? t : ((t < 5) ? t - 2 : 2);
      Aq = &sG[pa][0][0]; Bq = &sG[pb][0][0];
    } else if (ch < 13) { Aq = &sP[ch - 9][0][0]; Bq = nullptr; }
    else {
      const int k = ch - 13;
      Aq = &sG[k >> 2][0][0]; Bq = &sP[k & 3][0][0];
    }
    blur7x7(Aq, Bq, sH[wave], sS[ch], colM, half, lane);
  }
  __syncthreads();

  // Per-pixel: var_I + eps*I, cov_Ip, symmetric 3x3 inverse (Cramer), a & b
  float s[25];
#pragma unroll
  for (int i = 0; i < 25; ++i) s[i] = sS[i][tid];

  const float mI0 = s[0], mI1 = s[1], mI2 = s[2];
  const float m00 = s[3] - mI0 * mI0 + EPS;
  const float m01 = s[4] - mI0 * mI1;
  const float m02 = s[5] - mI0 * mI2;
  const float m11 = s[6] - mI1 * mI1 + EPS;
  const float m12 = s[7] - mI1 * mI2;
  const float m22 = s[8] - mI2 * mI2 + EPS;
  const float i00 = m11 * m22 - m12 * m12;
  const float i01 = m02 * m12 - m01 * m22;
  const float i02 = m01 * m12 - m02 * m11;
  const float i11 = m00 * m22 - m02 * m02;
  const float i12 = m01 * m02 - m00 * m12;
  const float i22 = m00 * m11 - m01 * m01;
  const float rdet = 1.0f / (m00 * i00 + m01 * i01 + m02 * i02);

  const int ty = tid >> 4, tx = tid & 15;
  const long long pix = ((long long)batch * HH + (y0 + ty)) * WW + (x0 + tx);
  const long long NP  = (long long)NB * HH * WW;

#pragma unroll
  for (int i = 0; i < 4; ++i) {
    const float mp = s[9 + i];
    const float c0 = s[13 + 0 + i] - mI0 * mp;
    const float c1 = s[13 + 4 + i] - mI1 * mp;
    const float c2 = s[13 + 8 + i] - mI2 * mp;
    const float a0 = rdet * (i00 * c0 + i01 * c1 + i02 * c2);
    const float a1 = rdet * (i01 * c0 + i11 * c1 + i12 * c2);
    const float a2 = rdet * (i02 * c0 + i12 * c1 + i22 * c2);
    const float bb = mp - (mI0 * a0 + mI1 * a1 + mI2 * a2);
    ab[(0 * 4 + i) * NP + pix] = a0;
    ab[(1 * 4 + i) * NP + pix] = a1;
    ab[(2 * 4 + i) * NP + pix] = a2;
    ab[(12 + i)    * NP + pix] = bb;
  }
}

// ---------------- Kernel 2: blur a,b (16 ch) + final combine --------------
__global__ __launch_bounds__(256) void gf_apply_kernel(
    const float* __restrict__ G, const float* __restrict__ ab,
    float* __restrict__ out) {
  __shared__ float sR[16][IT][TS];
  __shared__ float sH[8][24][16];
  __shared__ float sS[16][256];

  const int tid  = threadIdx.x;
  const int wave = __builtin_amdgcn_readfirstlane(tid >> 5);
  const int lane = tid & 31;
  const int colM = lane & 15, half = lane >> 4;
  const int x0 = blockIdx.x * 16, y0 = blockIdx.y * 16, batch = blockIdx.z;
  const long long NP = (long long)NB * HH * WW;

  const int ty = tid >> 4, tx = tid & 15;
  const int gy = y0 + ty, gx = x0 + tx;
  // Warm caches for the late per-pixel guidance reads (global_prefetch_b8).
#pragma unroll
  for (int c = 0; c < 3; ++c)
    __builtin_prefetch(&G[((batch * 3 + c) * HH + gy) * WW + gx], 0, 0);

  for (int idx = tid; idx < 16 * IT * 24; idx += 256) {
    const int ch  = idx / (IT * 24);
    const int rem = idx - ch * (IT * 24);
    const int r = rem / 24, c = rem - r * 24;
    float v = 0.0f;
    if (c < IT) {
      const int ry = refl(y0 - 3 + r), rx = refl(x0 - 3 + c);
      v = ab[(long long)ch * NP + ((long long)batch * HH + ry) * WW + rx];
    }
    sR[ch][r][c] = v;
  }
  sH[wave][22 + half][colM] = 0.0f;
  __syncthreads();

  for (int ch = wave; ch < 16; ch += 8)
    blur7x7(&sR[ch][0][0], nullptr, sH[wave], sS[ch], colM, half, lane);
  __syncthreads();

  const float g0 = G[((batch * 3 + 0) * HH + gy) * WW + gx];
  const float g1 = G[((batch * 3 + 1) * HH + gy) * WW + gx];
  const float g2 = G[((batch * 3 + 2) * HH + gy) * WW + gx];
#pragma unroll
  for (int i = 0; i < 4; ++i) {
    const float v = sS[12 + i][tid] + g0 * sS[i][tid] + g1 * sS[4 + i][tid] +
                    g2 * sS[8 + i][tid];
    out[(((long long)batch * 4 + i) * HH + gy) * WW + gx] = v;
  }
}

extern "C" void kernel_launch(void* const* d_in, const int* in_sizes, int n_in,
                              void* d_out, int out_size, void* d_ws,
                              size_t ws_size, hipStream_t stream) {
  (void)in_sizes; (void)n_in; (void)out_size; (void)ws_size;
  const float* G = (const float*)d_in[0];   // guidance [4,3,512,512]
  const float* P = (const float*)d_in[1];   // input    [4,4,512,512]
  float* abuf = (float*)d_ws;               // 16 planes x B*H*W (64 MB, L2-resident)
  float* out  = (float*)d_out;              // [4,4,512,512]

  dim3 grid(WW / 16, HH / 16, NB);
  gf_stats_kernel<<<grid, 256, 0, stream>>>(G, P, abuf);
  gf_apply_kernel<<<grid, 256, 0, stream>>>(G, abuf, out);
}